// AdultConnectome_26474178412844
// MI455X (gfx1250) — compile-verified
//
#include <hip/hip_runtime.h>
#include <stdint.h>

// ---------------------------------------------------------------------------
// Repeated COO SpMM: y <- A @ y, 4 layers. N=100000, E=3200000, B=16.
// Roofline: 410 MFLOP total (trivial); entirely L2-resident data movement
// (y ping-pong 2x6.4MB + indices/weights ~38MB << 192MB L2). Bottleneck is
// random 16B gathers + 51.2M f32 atomics per layer through L2.
// CDNA5 paths: async global->LDS staging of edge streams (ASYNCcnt,
// double-buffered, b64 per lane), 128-bit gathers, native f32 atomics.
// WMMA is mathematically inapplicable (no dense 16xK blocks at 3.2e-4
// density; MMA requires a shared B operand across rows).
// ---------------------------------------------------------------------------

#ifndef __has_builtin
#define __has_builtin(x) 0
#endif

#define BATCH 16
#define WAVES_PER_BLOCK 8
#define CHUNK 64  // edges staged per wave per chunk (2 per lane, b64)

__device__ __forceinline__ uint32_t lds_off(const void* p) {
  // Generic pointers to LDS on CDNA5: addr[31:0] is the LDS byte offset
  // (flat aperture, ISA 10.2).
  return (uint32_t)(uintptr_t)p;
}

__device__ __forceinline__ void async_b64_to_lds(uint32_t loff, const void* g) {
  asm volatile("global_load_async_to_lds_b64 %0, %1, off"
               :: "v"(loff), "v"((uint64_t)g)
               : "memory");
}

__device__ __forceinline__ void wait_async_le3() {
#if __has_builtin(__builtin_amdgcn_s_wait_asynccnt)
  __builtin_amdgcn_s_wait_asynccnt(3);
#else
  asm volatile("s_wait_asynccnt 0x3" ::: "memory");
#endif
}

__device__ __forceinline__ void wait_async_0() {
#if __has_builtin(__builtin_amdgcn_s_wait_asynccnt)
  __builtin_amdgcn_s_wait_asynccnt(0);
#else
  asm volatile("s_wait_asynccnt 0x0" ::: "memory");
#endif
}

__device__ __forceinline__ void wait_ds_0() {
  asm volatile("s_wait_dscnt 0x0" ::: "memory");
}

__device__ __forceinline__ void atomic_add_f32(float* p, float v) {
  (void)__hip_atomic_fetch_add(p, v, __ATOMIC_RELAXED, __HIP_MEMORY_SCOPE_AGENT);
}

__global__ void __launch_bounds__(256)
connectome_zero_kernel(float4* __restrict__ p, int n4) {
  int i = blockIdx.x * blockDim.x + threadIdx.x;
  if (i < n4) p[i] = make_float4(0.f, 0.f, 0.f, 0.f);
}

__global__ void __launch_bounds__(256)
connectome_spmm_kernel(const float* __restrict__ yin,
                       float* __restrict__ yout,
                       const float* __restrict__ wgt,
                       const int* __restrict__ row,
                       const int* __restrict__ col,
                       int E) {
  __shared__ alignas(16) int   s_col[2][WAVES_PER_BLOCK][CHUNK];
  __shared__ alignas(16) int   s_row[2][WAVES_PER_BLOCK][CHUNK];
  __shared__ alignas(16) float s_w  [2][WAVES_PER_BLOCK][CHUNK];

  const int lane = threadIdx.x & 31;
  const int wv   = threadIdx.x >> 5;
  const int gw   = blockIdx.x * WAVES_PER_BLOCK + wv;
  const int nw   = gridDim.x * WAVES_PER_BLOCK;
  const int nch  = (E + CHUNK - 1) / CHUNK;

  const int sub = lane >> 2;  // edge-in-pass (0..7)
  const int q   = lane & 3;   // which float4 quarter of the 16-wide row

  auto stage = [&](int ch, int b) {
    // Each lane stages 2 consecutive edges (8B) per array.
    int ei = ch * CHUNK + (lane << 1);
    if (ei > E - 2) ei = E - 2;  // tail clamp; duplicates never consumed
    // Async LDS writes are unordered vs. our earlier DS reads: drain DS
    // before re-staging a buffer we just consumed (WAR hazard).
    wait_ds_0();
    async_b64_to_lds(lds_off(&s_col[b][wv][lane << 1]), col + ei);
    async_b64_to_lds(lds_off(&s_row[b][wv][lane << 1]), row + ei);
    async_b64_to_lds(lds_off(&s_w  [b][wv][lane << 1]), wgt + ei);
  };

  int ch = gw;
  int b  = 0;
  if (ch < nch) stage(ch, 0);

  while (ch < nch) {
    const int  nxt  = ch + nw;
    const bool more = nxt < nch;
    if (more) {
      stage(nxt, b ^ 1);
      wait_async_le3();   // 3 newer ops outstanding -> current chunk is done
    } else {
      wait_async_0();
    }

#pragma unroll
    for (int p = 0; p < CHUNK / 8; ++p) {
      const int el = (p << 3) + sub;
      const int e  = ch * CHUNK + el;
      if (e < E) {
        const int   c = s_col[b][wv][el];
        const int   r = s_row[b][wv][el];
        const float t = s_w  [b][wv][el];
        const float4 v =
            *(const float4*)(yin + (size_t)c * BATCH + (q << 2));
        float* dst = yout + (size_t)r * BATCH + (q << 2);
        atomic_add_f32(dst + 0, t * v.x);
        atomic_add_f32(dst + 1, t * v.y);
        atomic_add_f32(dst + 2, t * v.z);
        atomic_add_f32(dst + 3, t * v.w);
      }
    }

    b ^= 1;
    ch = nxt;
  }
}

extern "C" void kernel_launch(void* const* d_in, const int* in_sizes, int n_in,
                              void* d_out, int out_size, void* d_ws, size_t ws_size,
                              hipStream_t stream) {
  const float* x   = (const float*)d_in[0];
  const float* w   = (const float*)d_in[1];
  const int*   row = (const int*)d_in[2];
  const int*   col = (const int*)d_in[3];
  const int    E   = in_sizes[1];
  const int    NB  = out_size;  // N_NODES * BATCH floats

  float* out = (float*)d_out;
  float* ws0 = (float*)d_ws;    // one ping-pong buffer (NB floats)

  const int n4    = NB / 4;
  const int zgrid = (n4 + 255) / 256;
  const int sgrid = 2048;       // 16384 waves: deep MLP for L2 latency hiding

  // layer 1: x -> ws0
  connectome_zero_kernel<<<zgrid, 256, 0, stream>>>((float4*)ws0, n4);
  connectome_spmm_kernel<<<sgrid, 256, 0, stream>>>(x, ws0, w, row, col, E);
  // layer 2: ws0 -> out
  connectome_zero_kernel<<<zgrid, 256, 0, stream>>>((float4*)out, n4);
  connectome_spmm_kernel<<<sgrid, 256, 0, stream>>>(ws0, out, w, row, col, E);
  // layer 3: out -> ws0
  connectome_zero_kernel<<<zgrid, 256, 0, stream>>>((float4*)ws0, n4);
  connectome_spmm_kernel<<<sgrid, 256, 0, stream>>>(out, ws0, w, row, col, E);
  // layer 4: ws0 -> out
  connectome_zero_kernel<<<zgrid, 256, 0, stream>>>((float4*)out, n4);
  connectome_spmm_kernel<<<sgrid, 256, 0, stream>>>(ws0, out, w, row, col, E);
}